// get_local_area_new_66743791780162
// MI455X (gfx1250) — compile-verified
//
#include <hip/hip_runtime.h>
#include <hip/hip_bf16.h>

#define NPTS   8192
#define NPOINT 2048
#define NSAMP  32

typedef __attribute__((ext_vector_type(2))) float v2f;
typedef __attribute__((ext_vector_type(8))) float v8f;

// ---------------------------------------------------------------------------
// Kernel 1: Furthest Point Sampling. One workgroup per batch (serial scan).
// xyz cached in LDS (96 KB of the 320 KB WGP LDS). 1024 thr x 8 pts each.
// Argmax via packed u64 key = (bits(min_d) << 32) | (8191 - idx)  -> max key
// gives max distance, ties resolved to smallest index (matches jnp.argmax).
// ---------------------------------------------------------------------------
__global__ __launch_bounds__(1024) void fps_kernel(const float* __restrict__ xyz,
                                                   int* __restrict__ fps_idx,
                                                   float* __restrict__ new_xyz) {
  __shared__ float sx[NPTS], sy[NPTS], sz[NPTS];
  __shared__ unsigned long long wred[32];
  __shared__ int s_winner;
  __shared__ int s_sel[NPOINT];

  const int b   = blockIdx.x;
  const int tid = threadIdx.x;
  const float* base = xyz + (size_t)b * NPTS * 3;

  for (int p = tid; p < NPTS; p += 1024) {
    sx[p] = base[3 * p + 0];
    sy[p] = base[3 * p + 1];
    sz[p] = base[3 * p + 2];
  }
  __syncthreads();

  float mind[8];
#pragma unroll
  for (int j = 0; j < 8; ++j) mind[j] = 1e10f;

  int last = 0;
  if (tid == 0) s_sel[0] = 0;

  for (int it = 1; it < NPOINT; ++it) {
    const float lx = sx[last], ly = sy[last], lz = sz[last];
    unsigned long long bestkey = 0ull;
#pragma unroll
    for (int j = 0; j < 8; ++j) {
      const int p = tid + j * 1024;
      const float dx = sx[p] - lx, dy = sy[p] - ly, dz = sz[p] - lz;
      const float d = dx * dx + dy * dy + dz * dz;
      const float m = fminf(mind[j], d);
      mind[j] = m;
      const unsigned long long key =
          ((unsigned long long)__float_as_uint(m) << 32) |
          (unsigned)(NPTS - 1 - p);
      bestkey = (key > bestkey) ? key : bestkey;
    }
    // wave32 max-reduce
#pragma unroll
    for (int off = 16; off > 0; off >>= 1) {
      const unsigned long long o = __shfl_xor(bestkey, off, 32);
      bestkey = (o > bestkey) ? o : bestkey;
    }
    if ((tid & 31) == 0) wred[tid >> 5] = bestkey;
    __syncthreads();
    if (tid < 32) {
      unsigned long long k2 = wred[tid];
#pragma unroll
      for (int off = 16; off > 0; off >>= 1) {
        const unsigned long long o = __shfl_xor(k2, off, 32);
        k2 = (o > k2) ? o : k2;
      }
      if (tid == 0) s_winner = NPTS - 1 - (int)(unsigned)(k2 & 0xFFFFFFFFu);
    }
    __syncthreads();
    last = s_winner;
    if (tid == 0) s_sel[it] = last;
  }
  __syncthreads();

  for (int q = tid; q < NPOINT; q += 1024) {
    const int si = s_sel[q];
    fps_idx[b * NPOINT + q] = si;
    float* o = new_xyz + ((size_t)b * NPOINT + q) * 3;
    o[0] = sx[si]; o[1] = sy[si]; o[2] = sz[si];
  }
}

// ---------------------------------------------------------------------------
// Kernel 2: kNN via V_WMMA_F32_16X16X4_F32, software-pipelined, unroll x2.
// One wave = 16 queries.  A(16x4) rows = (px,py,pz,|p|^2), B(4x16) cols =
// (-2qx,-2qy,-2qz,1), C = |q|^2 splat  =>  D[m][n] = ||p_m - q_n||^2 in the
// reference's |q|^2+|p|^2-2qp decomposition.  Per iteration we issue the two
// NEXT chunk loads (clamped addresses keep EXEC all-ones for WMMA), then run
// two independent WMMAs on resident data and insert candidates while the
// loads fly.  Lane n / lane n+16 each keep a top-32 (replace-max) list in
// LDS for query n over disjoint candidate halves; merged by selection sort.
// ---------------------------------------------------------------------------
__device__ __forceinline__ void topk_insert(unsigned long long* lists, int lane,
                                            const v8f& D, int pbase,
                                            unsigned long long& curMax,
                                            int& curPos) {
#pragma unroll
  for (int r = 0; r < 8; ++r) {
    const float d = D[r];
    unsigned int db = __float_as_uint(d);
    db = (db >> 31) ? ~db : (db | 0x80000000u);  // monotone float->u32
    const unsigned long long key =
        ((unsigned long long)db << 32) | (unsigned)(pbase + r);
    if (key < curMax) {
      lists[curPos * 32 + lane] = key;
      unsigned long long mx = 0ull; int mp = 0;
#pragma unroll
      for (int i = 0; i < 32; ++i) {
        const unsigned long long v = lists[i * 32 + lane];
        if (v > mx) { mx = v; mp = i; }
      }
      curMax = mx; curPos = mp;
    }
  }
}

__global__ __launch_bounds__(32) void knn_kernel(const float* __restrict__ xyz,
                                                 const int* __restrict__ fps_idx,
                                                 int* __restrict__ knn_idx) {
  __shared__ unsigned long long lists[32 * 32];  // lists[i*32 + lane]

  const int lane = threadIdx.x;
  const int b    = blockIdx.x >> 7;     // 128 query-groups per batch
  const int g    = blockIdx.x & 127;
  const int qcol = lane & 15;
  const int hi   = lane >> 4;
  const int q    = g * 16 + qcol;

  const float* P = xyz + (size_t)b * NPTS * 3;
  const int fq = fps_idx[b * NPOINT + q];
  const float qx = P[3 * fq + 0], qy = P[3 * fq + 1], qz = P[3 * fq + 2];

  v2f Bop;
  Bop.x = hi ? (-2.0f * qz) : (-2.0f * qx);
  Bop.y = hi ? 1.0f : (-2.0f * qy);
  const float cq = qx * qx + qy * qy + qz * qz;
  v8f Cop;
#pragma unroll
  for (int i = 0; i < 8; ++i) Cop[i] = cq;

#pragma unroll
  for (int i = 0; i < 32; ++i) lists[i * 32 + lane] = ~0ull;
  unsigned long long curMax = ~0ull;
  int curPos = 0;

  // Software pipeline: two 16-point chunks in flight.
  float p0x = P[3 * qcol + 0], p0y = P[3 * qcol + 1], p0z = P[3 * qcol + 2];
  float p1x = P[3 * (16 + qcol) + 0], p1y = P[3 * (16 + qcol) + 1],
        p1z = P[3 * (16 + qcol) + 2];

  for (int bas = 0; bas < NPTS; bas += 32) {
    // issue next-iteration loads first (clamped: uniform control flow)
    const int m0 = min(bas + 32 + qcol, NPTS - 1);
    const int m1 = min(bas + 48 + qcol, NPTS - 1);
    const float n0x = P[3 * m0 + 0], n0y = P[3 * m0 + 1], n0z = P[3 * m0 + 2];
    const float n1x = P[3 * m1 + 0], n1y = P[3 * m1 + 1], n1z = P[3 * m1 + 2];

    v2f A0, A1;
    A0.x = hi ? p0z : p0x;
    A0.y = hi ? (p0x * p0x + p0y * p0y + p0z * p0z) : p0y;
    A1.x = hi ? p1z : p1x;
    A1.y = hi ? (p1x * p1x + p1y * p1y + p1z * p1z) : p1y;

    v8f D0 = __builtin_amdgcn_wmma_f32_16x16x4_f32(
        false, A0, false, Bop, (short)0, Cop, false, false);
    v8f D1 = __builtin_amdgcn_wmma_f32_16x16x4_f32(
        false, A1, false, Bop, (short)0, Cop, false, false);

    topk_insert(lists, lane, D0, bas + hi * 8, curMax, curPos);
    topk_insert(lists, lane, D1, bas + 16 + hi * 8, curMax, curPos);

    p0x = n0x; p0y = n0y; p0z = n0z;
    p1x = n1x; p1y = n1y; p1z = n1z;
  }
  __syncthreads();

  if (lane < 16) {
    unsigned long long usedMask = 0ull;
    const int outBase = (b * NPOINT + q) * NSAMP;
    for (int k = 0; k < NSAMP; ++k) {
      unsigned long long mn = ~0ull; int mi = 0;
      for (int i = 0; i < 64; ++i) {
        if ((usedMask >> i) & 1ull) continue;
        const int l = (i < 32) ? lane : (lane + 16);
        const unsigned long long v = lists[(i & 31) * 32 + l];
        if (v < mn) { mn = v; mi = i; }
      }
      usedMask |= (1ull << mi);
      knn_idx[outBase + k] = (int)(unsigned)(mn & 0xFFFFFFFFu);
    }
  }
}

// ---------------------------------------------------------------------------
// Kernel 3: gather + feature concat (bandwidth bound, ~410 MB stream out).
// One block per (b, q).  group_fts written with non-temporal stores (no reuse,
// footprint 2x the 192 MB L2).  points_fts (12.6 MB) stays L2-resident.
// ---------------------------------------------------------------------------
__global__ __launch_bounds__(256) void gather_kernel(const float* __restrict__ xyz,
                                                     const float* __restrict__ fts,
                                                     const int* __restrict__ fps_idx,
                                                     const int* __restrict__ knn_idx,
                                                     float* __restrict__ out) {
  float* out_gxyz = out;                         // (2,2048,32,3)
  float* out_gfts = out + 393216;                // (2,128,3,2048,32)
  float* out_nfts = out + 101068800;             // (2,128,3,2048,1)

  const int b = blockIdx.x >> 11;
  const int q = blockIdx.x & 2047;
  const int tid = threadIdx.x;

  __shared__ int sidx[NSAMP];
  __shared__ int sfq;
  if (tid < NSAMP) sidx[tid] = knn_idx[(b * NPOINT + q) * NSAMP + tid];
  if (tid == 0) sfq = fps_idx[b * NPOINT + q];
  __syncthreads();
  const int fq = sfq;

  if (tid < 96) {
    const int k = tid / 3, j = tid % 3;
    out_gxyz[(((size_t)b * NPOINT + q) * NSAMP + k) * 3 + j] =
        xyz[((size_t)b * NPTS + sidx[k]) * 3 + j];
  }

  const float* F = fts + (size_t)b * 64 * 3 * NPTS;
  for (int e = tid; e < 64 * 3 * NSAMP; e += 256) {
    const int k = e & 31;
    const int row = e >> 5;            // 0..191
    const int c = row / 3, d3 = row - 3 * c;
    const float* Frow = F + (size_t)(c * 3 + d3) * NPTS;
    const float center = Frow[fq];
    const float nbr = Frow[sidx[k]];
    const size_t o1 = ((size_t)((b * 128 + c) * 3 + d3) * NPOINT + q) * NSAMP + k;
    const size_t o2 = ((size_t)((b * 128 + c + 64) * 3 + d3) * NPOINT + q) * NSAMP + k;
    __builtin_nontemporal_store(nbr - center, out_gfts + o1);
    __builtin_nontemporal_store(center, out_gfts + o2);
    if (k == 0) {
      out_nfts[(size_t)((b * 128 + c) * 3 + d3) * NPOINT + q] = nbr - center;
      out_nfts[(size_t)((b * 128 + c + 64) * 3 + d3) * NPOINT + q] = center;
    }
  }
}

// ---------------------------------------------------------------------------
extern "C" void kernel_launch(void* const* d_in, const int* in_sizes, int n_in,
                              void* d_out, int out_size, void* d_ws, size_t ws_size,
                              hipStream_t stream) {
  const float* xyz = (const float*)d_in[0];   // (2, 8192, 3)
  const float* fts = (const float*)d_in[1];   // (2, 64, 3, 8192)
  float* out = (float*)d_out;

  int* fps_idx = (int*)d_ws;                  // 2*2048
  int* knn_idx = fps_idx + 2 * NPOINT;        // 2*2048*32

  // new_xyz region lives at offset 101056512 in the flat output
  fps_kernel<<<2, 1024, 0, stream>>>(xyz, fps_idx, out + 101056512);
  knn_kernel<<<2 * (NPOINT / 16), 32, 0, stream>>>(xyz, fps_idx, knn_idx);
  gather_kernel<<<2 * NPOINT, 256, 0, stream>>>(xyz, fts, fps_idx, knn_idx, out);
}